// Signature_74835510166228
// MI455X (gfx1250) — compile-verified
//
#include <hip/hip_runtime.h>

// Truncated path signature depth-6, B=128, L=256, C=8.
// Kernel A: levels 1..5 streaming Chen scan, full state in LDS (needs CDNA5's 320KB/WGP).
// Kernel B: level 6 reformulated as a 512x512x765 fp32 GEMM per batch using
//           V_WMMA_F32_16X16X4_F32 (rank-3-per-step factorization via Chen + suffix scans).

typedef __attribute__((ext_vector_type(2))) float v2f;
typedef __attribute__((ext_vector_type(8))) float v8f;

#define TT 255          // number of segments (Chen steps), state starts at 0
#define SIG_LEN 299592  // 8+64+512+4096+32768+262144
#define LVL6_OFF 37448
#define ASTR 50         // K-panel stride (floats), even -> 8B-aligned b64 LDS loads

// ---------------------------------------------------------------------------
// Kernel A: levels 1..5. One workgroup per batch, 256 threads.
// LDS: S5(32768) S4(4096) seg4(4096) seg3(512) S3(512) S2(64) S1(8) dx(2040)
// ---------------------------------------------------------------------------
__global__ __launch_bounds__(256, 1)
void sig_levels15(const float* __restrict__ path, float* __restrict__ out)
{
    extern __shared__ float sm[];
    float* S5  = sm;              // 32768
    float* S4  = S5  + 32768;     // 4096
    float* sg4 = S4  + 4096;      // 4096
    float* sg3 = sg4 + 4096;      // 512
    float* S3  = sg3 + 512;       // 512
    float* S2  = S3  + 512;       // 64
    float* S1  = S2  + 64;        // 8
    float* dxs = S1  + 8;         // 2040

    const int tid = threadIdx.x;
    const int b   = blockIdx.x;
    const float* pb = path + (size_t)b * 256 * 8;

    for (int i = tid; i < TT * 8; i += 256) dxs[i] = pb[i + 8] - pb[i];
    for (int i = tid; i < 32768;  i += 256) S5[i] = 0.f;
    for (int i = tid; i < 4096;   i += 256) S4[i] = 0.f;
    for (int i = tid; i < 512;    i += 256) S3[i] = 0.f;
    if (tid < 64) S2[tid] = 0.f;
    if (tid < 8)  S1[tid] = 0.f;
    __syncthreads();

    for (int t = 0; t < TT; ++t) {
        float d[8];
#pragma unroll
        for (int c = 0; c < 8; ++c) d[c] = dxs[t * 8 + c];

        for (int j = tid; j < 512; j += 256)                 // seg3 = d^3/6
            sg3[j] = d[(j >> 6) & 7] * d[(j >> 3) & 7] * d[j & 7] * (1.f / 6.f);
        __syncthreads();
        for (int j = tid; j < 4096; j += 256)                // seg4 = seg3 (x) d / 4
            sg4[j] = sg3[j >> 3] * d[j & 7] * 0.25f;
        __syncthreads();

        // S5 += seg5 + S1(x)seg4 + S2(x)seg3 + S3(x)seg2 + S4(x)d   (pre-step states)
        for (int g = tid; g < 4096; g += 256) {
            float a4 = sg4[g];
            float c0 = a4 * 0.2f + S4[g];                    // (seg5/d + S4) factor
            float s3h = S3[g >> 3] * d[g & 7] * 0.5f;        // S3 * seg2 head
            float s2  = S2[g >> 6];
            float s1  = S1[g >> 9];
            const int b3 = (g & 63) << 3;                    // seg3[(i3,i4,i5)] base
            const int b4 = (g & 511) << 3;                   // seg4[(i2..i5)] base
#pragma unroll
            for (int i5 = 0; i5 < 8; ++i5) {
                float inc = (c0 + s3h) * d[i5] + s2 * sg3[b3 + i5] + s1 * sg4[b4 + i5];
                S5[(g << 3) | i5] += inc;
            }
        }
        __syncthreads();
        for (int j = tid; j < 4096; j += 256) {
            float inc = sg4[j] + S1[j >> 9] * sg3[j & 511]
                      + S2[j >> 6] * (d[(j >> 3) & 7] * d[j & 7] * 0.5f)
                      + S3[j >> 3] * d[j & 7];
            S4[j] += inc;
        }
        __syncthreads();
        for (int j = tid; j < 512; j += 256)
            S3[j] += sg3[j] + S1[j >> 6] * (d[(j >> 3) & 7] * d[j & 7] * 0.5f)
                   + S2[j >> 3] * d[j & 7];
        __syncthreads();
        if (tid < 64) S2[tid] += d[tid >> 3] * d[tid & 7] * 0.5f + S1[tid >> 3] * d[tid & 7];
        __syncthreads();
        if (tid < 8) S1[tid] += d[tid];
        __syncthreads();
    }

    float* ob = out + (size_t)b * SIG_LEN;
    if (tid < 8)  ob[tid] = S1[tid];
    if (tid < 64) ob[8 + tid] = S2[tid];
    for (int j = tid; j < 512;   j += 256) ob[72 + j]   = S3[j];
    for (int j = tid; j < 4096;  j += 256) ob[584 + j]  = S4[j];
    for (int j = tid; j < 32768; j += 256) ob[4680 + j] = S5[j];
}

// ---------------------------------------------------------------------------
// Kernel B: level 6 as S6 = sum_t (U_t(x)B1_t + V_t(x)B2_t + W_t(x)B3_t),
// a 512x512 x (3*255) fp32 GEMM per batch computed with v_wmma_f32_16x16x4_f32.
// Grid (4,128): blockIdx.x selects 256x256 quadrant, blockIdx.y = batch.
// 512 threads = 16 wave32; each wave holds a 16x256 strip (16 v8f accumulators).
// ---------------------------------------------------------------------------
__global__ __launch_bounds__(512, 1)
void sig_level6(const float* __restrict__ path, float* __restrict__ out)
{
    extern __shared__ float sm[];
    float* Ast = sm;                    // 256*ASTR  A panel [m][k], k=3*step+{U,V,W}
    float* Bst = Ast + 256 * ASTR;      // 256*ASTR  B panel [n][k], k=3*step+{B1,B2,B3}
    float* Gt  = Bst + 256 * ASTR;      // 255*64    suffix E+F tables
    float* Dt  = Gt + TT * 64;          // 255*8     suffix sum of d
    float* dxs = Dt + TT * 8;           // 255*8
    float* S2s = dxs + TT * 8;          // 64

    const int tid  = threadIdx.x;
    const int b    = blockIdx.y;
    const int m0   = (blockIdx.x >> 1) * 256;
    const int n0   = (blockIdx.x & 1) * 256;
    const int lane = tid & 31;
    const int wave = tid >> 5;

    const float* pb = path + (size_t)b * 2048;
    for (int i = tid; i < TT * 8; i += 512) dxs[i] = pb[i + 8] - pb[i];
    if (tid < 64) S2s[tid] = 0.f;
    __syncthreads();
    if (tid < 8) {                       // D_t = sum_{s>t} d_s
        float a = 0.f; Dt[254 * 8 + tid] = 0.f;
        for (int t = 253; t >= 0; --t) { a += dxs[(t + 1) * 8 + tid]; Dt[t * 8 + tid] = a; }
    }
    __syncthreads();
    if (tid < 64) {                      // G_t = E_t + F_t = sum_{s>t} d_s[b]*(D_s[c] + d_s[c]/2)
        const int qb = tid >> 3, qc = tid & 7;
        float a = 0.f; Gt[254 * 64 + tid] = 0.f;
        for (int t = 253; t >= 0; --t) {
            float db = dxs[(t + 1) * 8 + qb];
            a += db * (Dt[(t + 1) * 8 + qc] + 0.5f * dxs[(t + 1) * 8 + qc]);
            Gt[t * 64 + tid] = a;
        }
    }

    const bool isA = tid < 256;
    const int  idx = isA ? tid : (tid - 256);
    const int  e1 = (idx >> 6) & 7, e2 = (idx >> 3) & 7, e3 = idx & 7;
    const int  p12 = idx >> 3;

    float S1r[8]; float S3p = 0.f;
#pragma unroll
    for (int c = 0; c < 8; ++c) S1r[c] = 0.f;
    v8f acc[16];
    const v8f vzero = {0.f, 0.f, 0.f, 0.f, 0.f, 0.f, 0.f, 0.f};
#pragma unroll
    for (int i = 0; i < 16; ++i) acc[i] = vzero;

    for (int chunk = 0; chunk < 16; ++chunk) {
        const int base = chunk * 16;
        for (int sl = 0; sl < 16; ++sl) {
            __syncthreads();             // staging free (prev GEMM done) + states stable
            const int t = base + sl;
            float dv[8];
            if (t < TT) {
#pragma unroll
                for (int c = 0; c < 8; ++c) dv[c] = dxs[t * 8 + c];
                const float d1 = dv[e1], d2 = dv[e2], d3 = dv[e3];
                if (isA) {
                    float g3  = d1 * d2 * d3;                 // 6*seg3[p]
                    float x12 = S1r[e1] * (d2 * d3 * 0.5f);   // S1 (x) seg2
                    float x21 = S2s[p12] * d3;                // S2 (x) d
                    float U = g3 * (1.f / 120.f) + x12 * 0.1f        + x21 * 0.25f       + S3p;
                    float V = g3 * (1.f / 60.f)  + x12 * (1.f / 6.f) + x21 * (1.f / 3.f) + S3p;
                    float W = g3 * (1.f / 24.f)  + x12 * (1.f / 3.f) + x21 * 0.5f        + S3p;
                    S3p += g3 * (1.f / 6.f) + x12 + x21;      // Chen level-3 row update
                    float* ar = &Ast[idx * ASTR + sl * 3];
                    ar[0] = U; ar[1] = V; ar[2] = W;
                } else {
                    float B1 = d1 * d2 * d3 * (1.f / 6.f);              // seg3[q]
                    float B2 = d1 * d2 * 0.5f * Dt[t * 8 + e3];         // seg2*suffix(d)
                    float B3 = d1 * Gt[t * 64 + (idx & 63)];            // d*(E+F)
                    float* br = &Bst[idx * ASTR + sl * 3];
                    br[0] = B1; br[1] = B2; br[2] = B3;
                }
            } else {                      // pad last K-panel with zeros
                float* r = isA ? &Ast[idx * ASTR + sl * 3] : &Bst[idx * ASTR + sl * 3];
                r[0] = 0.f; r[1] = 0.f; r[2] = 0.f;
            }
            __syncthreads();              // all reads of S2s done; now advance states
            if (t < TT) {
                if (tid < 64)
                    S2s[tid] += dv[tid >> 3] * dv[tid & 7] * 0.5f + S1r[tid >> 3] * dv[tid & 7];
#pragma unroll
                for (int c = 0; c < 8; ++c) S1r[c] += dv[c];
            }
        }
        __syncthreads();                  // K=48 panel staged

        // rank-48 update: 16 tiles/wave x 12 wmma (K step 4)
        const int arow = wave * 16 + (lane & 15);
        const int koff = (lane < 16) ? 0 : 2;
#pragma unroll
        for (int kk = 0; kk < 48; kk += 4) {
            v2f af = *(const v2f*)&Ast[arow * ASTR + kk + koff];
#pragma unroll
            for (int tn = 0; tn < 16; ++tn) {
                v2f bf = *(const v2f*)&Bst[(tn * 16 + (lane & 15)) * ASTR + kk + koff];
                acc[tn] = __builtin_amdgcn_wmma_f32_16x16x4_f32(
                    false, af, false, bf, (short)0, acc[tn], false, false);
            }
        }
    }

    // Writeback: VGPR r holds M=r (lanes 0-15) / M=r+8 (lanes 16-31), N=lane&15.
    float* o6 = out + (size_t)b * SIG_LEN + LVL6_OFF;
    const int mrow = m0 + wave * 16 + ((lane >> 4) << 3);
    const int ncol = n0 + (lane & 15);
    for (int tn = 0; tn < 16; ++tn) {
#pragma unroll
        for (int r = 0; r < 8; ++r)
            o6[(size_t)(mrow + r) * 512 + ncol + tn * 16] = acc[tn][r];
    }
}

// ---------------------------------------------------------------------------
extern "C" void kernel_launch(void* const* d_in, const int* in_sizes, int n_in,
                              void* d_out, int out_size, void* d_ws, size_t ws_size,
                              hipStream_t stream) {
    (void)in_sizes; (void)n_in; (void)out_size; (void)d_ws; (void)ws_size;
    const float* path = (const float*)d_in[0];   // [128,256,8] fp32; d_in[1]=depth==6
    float* out = (float*)d_out;                  // [128, 299592] fp32

    const size_t ldsA = (size_t)(32768 + 4096 + 4096 + 512 + 512 + 64 + 8 + TT * 8) * sizeof(float);
    const size_t ldsB = (size_t)(256 * ASTR * 2 + TT * 64 + TT * 8 + TT * 8 + 64) * sizeof(float);
    // CDNA5 allows up to 320KB LDS per workgroup; raise the dynamic-LDS cap.
    hipFuncSetAttribute((const void*)sig_levels15,
                        hipFuncAttributeMaxDynamicSharedMemorySize, (int)ldsA);
    hipFuncSetAttribute((const void*)sig_level6,
                        hipFuncAttributeMaxDynamicSharedMemorySize, (int)ldsB);

    sig_levels15<<<dim3(128), dim3(256), ldsA, stream>>>(path, out);
    sig_level6<<<dim3(4, 128), dim3(512), ldsB, stream>>>(path, out);
}